// FocalLossWithVariencePenalty_50259707298726
// MI455X (gfx1250) — compile-verified
//
#include <hip/hip_runtime.h>
#include <hip/hip_bf16.h>

typedef __attribute__((ext_vector_type(2))) float v2f;
typedef __attribute__((ext_vector_type(4))) float f4;
typedef __attribute__((ext_vector_type(8))) float v8f;

#define BATCH 16
#define CHAN 8
#define HW (512*512)
#define HW4 (HW/4)                              // 65536 float4 per channel
#define BLOCKS_PER_BATCH 128
#define THREADS 256
#define F4_PER_BLOCK (HW4 / BLOCKS_PER_BATCH)   // 512
#define ITERS (F4_PER_BLOCK / THREADS)          // 2
#define NBLOCKS (BATCH * BLOCKS_PER_BATCH)      // 2048
#define LN2F 0.69314718055994530942f

__device__ __forceinline__ float bce_term(float p, float tv) {
    // t*max(log p,-100) + (1-t)*max(log(1-p),-100), log via v_log_f32
    float lp = fmaxf(__log2f(p) * LN2F, -100.0f);
    float l1 = fmaxf(__log2f(1.0f - p) * LN2F, -100.0f);
    return fmaf(tv, lp - l1, l1);
}

// Kernel 1: single streaming pass. Block = (batch b, spatial slab). Walks all 8
// channels at the same spatial offsets so adjacent-channel dot products need no
// second read of `inputs`. Emits 16 partial sums per block (deterministic tree).
__global__ __launch_bounds__(THREADS)
void fused_pass(const float* __restrict__ X, const float* __restrict__ T,
                float* __restrict__ ws) {
    const int b   = blockIdx.y;
    const int blk = blockIdx.x;
    const int tid = threadIdx.x;
    const f4* xb = (const f4*)X + (size_t)b * CHAN * HW4;
    const f4* tb = (const f4*)T + (size_t)b * CHAN * HW4;

    float sq[CHAN];
    float dt[CHAN - 1];
    float bce = 0.0f;
#pragma unroll
    for (int c = 0; c < CHAN; ++c) sq[c] = 0.0f;
#pragma unroll
    for (int c = 0; c < CHAN - 1; ++c) dt[c] = 0.0f;

    const int base = blk * F4_PER_BLOCK;
#pragma unroll
    for (int it = 0; it < ITERS; ++it) {
        const int i = base + it * THREADS + tid;
        f4 prev = {};
#pragma unroll
        for (int c = 0; c < CHAN; ++c) {
            f4 xv = __builtin_nontemporal_load(&xb[(size_t)c * HW4 + i]);
            f4 tv = __builtin_nontemporal_load(&tb[(size_t)c * HW4 + i]);
            sq[c] += xv.x*xv.x + xv.y*xv.y + xv.z*xv.z + xv.w*xv.w;
            if (c > 0)
                dt[c-1] += prev.x*xv.x + prev.y*xv.y + prev.z*xv.z + prev.w*xv.w;
            prev = xv;
            bce += bce_term(xv.x, tv.x) + bce_term(xv.y, tv.y)
                 + bce_term(xv.z, tv.z) + bce_term(xv.w, tv.w);
        }
    }

    // 16 quantities: [sq0..7, dot0..6, bce]. Wave32 xor-tree then cross-wave LDS.
    float vals[16];
#pragma unroll
    for (int c = 0; c < 8; ++c) vals[c] = sq[c];
#pragma unroll
    for (int c = 0; c < 7; ++c) vals[8 + c] = dt[c];
    vals[15] = bce;

#pragma unroll
    for (int q = 0; q < 16; ++q) {
        float v = vals[q];
#pragma unroll
        for (int off = 16; off >= 1; off >>= 1) v += __shfl_xor(v, off, 32);
        vals[q] = v;
    }

    __shared__ float s_red[8 * 16];
    const int wave = tid >> 5, lane = tid & 31;
    if (lane == 0) {
#pragma unroll
        for (int q = 0; q < 16; ++q) s_red[wave * 16 + q] = vals[q];
    }
    __syncthreads();
    if (tid < 16) {
        float s = 0.0f;
#pragma unroll
        for (int w = 0; w < 8; ++w) s += s_red[w * 16 + tid];
        ws[((size_t)b * BLOCKS_PER_BATCH + blk) * 16 + tid] = s;
    }
}

// Kernel 2: deterministic finalize. Channel norms + dots by 1 thread each;
// the 2048 BCE partials reduced on wave 0 via V_WMMA_F32_16X16X4_F32 with a
// ones A-matrix (D[m][n] = sum_k B[k][n]), accumulator-chained in f32.
__global__ __launch_bounds__(256)
void finalize(const float* __restrict__ ws, float* __restrict__ out) {
    __shared__ float s_na[BATCH * CHAN];         // 128
    __shared__ float s_dot[BATCH * (CHAN - 1)];  // 112
    __shared__ float s_cos[BATCH * (CHAN - 1)];  // 112
    __shared__ float s_bce;
    const int tid = threadIdx.x;

    if (tid < BATCH * CHAN) {                    // per-channel ||x||
        const int bb = tid >> 3, cc = tid & 7;
        float s = 0.0f;
        for (int blk = 0; blk < BLOCKS_PER_BATCH; ++blk)
            s += ws[((size_t)bb * BLOCKS_PER_BATCH + blk) * 16 + cc];
        s_na[tid] = sqrtf(s);
    } else if (tid < BATCH * CHAN + BATCH * (CHAN - 1)) {   // adjacent dots
        const int pi = tid - BATCH * CHAN;
        const int bb = pi / 7, cc = pi % 7;
        float s = 0.0f;
        for (int blk = 0; blk < BLOCKS_PER_BATCH; ++blk)
            s += ws[((size_t)bb * BLOCKS_PER_BATCH + blk) * 16 + 8 + cc];
        s_dot[pi] = s;
    }

    if (tid < 32) {  // wave 0, EXEC all-ones: WMMA f32 reduction of 2048 partials
        v8f acc = {};
        v2f av; av.x = 1.0f; av.y = 1.0f;        // A = ones (16x4)
        const int n  = tid & 15;
        const int kb = (tid < 16) ? 0 : 2;       // B-matrix 4x16 VGPR layout
        for (int it = 0; it < NBLOCKS / 64; ++it) {
            const int base = it * 64;
            v2f bv;
            bv.x = ws[(size_t)(base + kb * 16 + n) * 16 + 15];
            bv.y = ws[(size_t)(base + (kb + 1) * 16 + n) * 16 + 15];
            acc = __builtin_amdgcn_wmma_f32_16x16x4_f32(
                false, av, false, bv, (short)0, acc, false, false);
        }
        float col = acc[0];                      // lane L: column-sum n = L%16
        float tot = 0.0f;
        for (int l = 0; l < 16; ++l) tot += __shfl(col, l, 32);
        if (tid == 0) s_bce = tot;
    }
    __syncthreads();

    if (tid < BATCH * (CHAN - 1)) {
        const int bb = tid / 7, cc = tid % 7;
        const float na = fmaxf(s_na[bb * 8 + cc],     1e-8f);
        const float nb = fmaxf(s_na[bb * 8 + cc + 1], 1e-8f);
        s_cos[tid] = s_dot[tid] / (na * nb);
    }
    __syncthreads();

    if (tid == 0) {
        float similar = 0.0f;
        for (int i = 0; i < BATCH * (CHAN - 1); ++i) similar += s_cos[i];
        float bce = -s_bce / (float)((size_t)BATCH * CHAN * HW);
        float e = 1.0f - expf(-bce);
        float focal = 0.5f * e * e * bce;
        out[0] = focal + similar * (2.0f / (BATCH * CHAN));
    }
}

extern "C" void kernel_launch(void* const* d_in, const int* in_sizes, int n_in,
                              void* d_out, int out_size, void* d_ws, size_t ws_size,
                              hipStream_t stream) {
    const float* X = (const float*)d_in[0];
    const float* T = (const float*)d_in[1];
    float* out = (float*)d_out;
    float* ws  = (float*)d_ws;   // needs 2048*16*4 = 128 KB

    dim3 grid(BLOCKS_PER_BATCH, BATCH);
    fused_pass<<<grid, THREADS, 0, stream>>>(X, T, ws);
    finalize<<<1, 256, 0, stream>>>(ws, out);
}